// SpGraphAttentionLayer_69758858822412
// MI455X (gfx1250) — compile-verified
//
#include <hip/hip_runtime.h>
#include <cmath>

#define IN_F  256
#define OUT_F 64
#define LRELU_ALPHA 0.2f

typedef __attribute__((ext_vector_type(2))) float v2f;
typedef __attribute__((ext_vector_type(8))) float v8f;

// ---------------------------------------------------------------------------
// h[N x 64] = x[N x 256] @ W[256 x 64], exact f32 via v_wmma_f32_16x16x4_f32.
// Block = 256 threads = 8 waves; each wave owns a 16-row strip and all 4
// 16x16 col tiles. W (64 KB) staged in LDS once per block, swizzled by K-pair
// so each B fragment is one contiguous ds_load_b64.
// Branch-free inner loop: A row clamped to N-1 (garbage rows never stored,
// since D row M depends only on A row M and stores are guarded by rr < N).
// ---------------------------------------------------------------------------
__global__ __launch_bounds__(256) void gat_gemm_wmma(
    const float* __restrict__ x, const float* __restrict__ W,
    float* __restrict__ h, int N)
{
    // Wlds[(k>>1)*128 + c*2 + (k&1)] = W[k*64 + c]  -> B frag = float2
    __shared__ float Wlds[IN_F * OUT_F];   // 64 KB of 320 KB WGP LDS
    const int tid = threadIdx.x;
    for (int i = tid; i < IN_F * OUT_F; i += 256) {
        const int k = i / OUT_F, c = i % OUT_F;
        Wlds[(k >> 1) * (2 * OUT_F) + c * 2 + (k & 1)] = W[i];
    }
    __syncthreads();

    const int wave = tid >> 5;
    const int lane = tid & 31;
    const int half = lane >> 4;            // 0: lanes 0-15, 1: lanes 16-31
    const int l16  = lane & 15;
    const int tile_row0 = blockIdx.x * 128 + wave * 16;
    const int arow = tile_row0 + l16;      // A-matrix row for this lane
    const int lrow = arow < N ? arow : (N - 1);   // clamp: no OOB, no branch

    // Per-lane base pointers (loop-invariant)
    const float* xptr = x + (size_t)lrow * IN_F + 2 * half;        // A pairs
    const float* bptr = Wlds + half * (2 * OUT_F) + l16 * 2;       // B pairs

    v8f acc[4] = {};                       // 4 x (16x16 f32) = 16 rows x 64 cols

    for (int k0 = 0; k0 < IN_F; k0 += 4) {
        const float2 t = *(const float2*)(xptr + k0);
        v2f a; a.x = t.x; a.y = t.y;
        const float* bk = bptr + (k0 >> 1) * (2 * OUT_F);
#pragma unroll
        for (int n = 0; n < 4; ++n) {
            const float2 u = *(const float2*)(bk + n * 32);        // ds_load_b64
            v2f b; b.x = u.x; b.y = u.y;
            acc[n] = __builtin_amdgcn_wmma_f32_16x16x4_f32(
                false, a, false, b, (short)0, acc[n], false, false);
        }
    }

    // D layout: elem r of acc[n] -> row tile_row0 + 8*half + r, col n*16+l16
    const int rbase = tile_row0 + 8 * half;
#pragma unroll
    for (int r = 0; r < 8; ++r) {
        const int rr = rbase + r;
        if (rr < N) {
#pragma unroll
            for (int n = 0; n < 4; ++n)
                h[(size_t)rr * OUT_F + n * 16 + l16] = acc[n][r];
        }
    }
}

// ---------------------------------------------------------------------------
// s_src[i] = h[i,:] . attn[0:64], s_dst[i] = h[i,:] . attn[64:128]
// One wave32 per node, 2 features per lane, butterfly reduce.
// ---------------------------------------------------------------------------
__global__ __launch_bounds__(256) void gat_scores(
    const float* __restrict__ h, const float* __restrict__ attn,
    float* __restrict__ s_src, float* __restrict__ s_dst, int N)
{
    const int node = (blockIdx.x * blockDim.x + threadIdx.x) >> 5;
    const int lane = threadIdx.x & 31;
    if (node >= N) return;                 // whole wave exits together
    const float* hr = h + (size_t)node * OUT_F;
    const float h0 = hr[lane], h1 = hr[lane + 32];
    float ps = h0 * attn[lane]         + h1 * attn[lane + 32];
    float pd = h0 * attn[OUT_F + lane] + h1 * attn[OUT_F + lane + 32];
    for (int off = 16; off > 0; off >>= 1) {
        ps += __shfl_down(ps, off, 32);
        pd += __shfl_down(pd, off, 32);
    }
    if (lane == 0) { s_src[node] = ps; s_dst[node] = pd; }
}

__device__ __forceinline__ float edge_logit(const float* s_src, const float* s_dst,
                                            int s, int d) {
    const float v = s_src[s] + s_dst[d];
    return -(v > 0.f ? v : LRELU_ALPHA * v);   // -leaky_relu(v, 0.2)
}

__global__ void gat_init(float* __restrict__ out, float* __restrict__ m,
                         float* __restrict__ denom, int N)
{
    const int i = blockIdx.x * blockDim.x + threadIdx.x;
    if (i < N * OUT_F) out[i] = 0.f;
    if (i < N) { m[i] = -INFINITY; denom[i] = 0.f; }
}

__global__ void gat_edge_max(const int* __restrict__ src, const int* __restrict__ dst,
                             const float* __restrict__ s_src, const float* __restrict__ s_dst,
                             float* __restrict__ m, int E)
{
    const int i = blockIdx.x * blockDim.x + threadIdx.x;
    if (i >= E) return;
    const int s = src[i];
    const float e = edge_logit(s_src, s_dst, s, dst[i]);
    atomicMax(&m[s], e);                   // global_atomic_max_num_f32
}

__global__ void gat_edge_sum(const int* __restrict__ src, const int* __restrict__ dst,
                             const float* __restrict__ s_src, const float* __restrict__ s_dst,
                             const float* __restrict__ m, float* __restrict__ denom, int E)
{
    const int i = blockIdx.x * blockDim.x + threadIdx.x;
    if (i >= E) return;
    const int s = src[i];
    const float e = edge_logit(s_src, s_dst, s, dst[i]);
    atomicAdd(&denom[s], __expf(e - m[s]));
}

// One wave32 per edge; lane covers features {lane, lane+32}. h[dst] gathers
// and the f32 atomic-adds are coalesced per edge and L2-resident (h = 25.6 MB).
__global__ __launch_bounds__(256) void gat_edge_aggr(
    const int* __restrict__ src, const int* __restrict__ dst,
    const float* __restrict__ s_src, const float* __restrict__ s_dst,
    const float* __restrict__ m, const float* __restrict__ denom,
    const float* __restrict__ h, float* __restrict__ out, int E)
{
    const int edge = (blockIdx.x * blockDim.x + threadIdx.x) >> 5;
    const int lane = threadIdx.x & 31;
    if (edge >= E) return;                 // wave-uniform exit
    const int s = src[edge], d = dst[edge];
    const float e = edge_logit(s_src, s_dst, s, d);
    const float w = __expf(e - m[s]) / denom[s];
    const float* hr = h + (size_t)d * OUT_F;
    float* orow = out + (size_t)s * OUT_F;
    atomicAdd(&orow[lane],      w * hr[lane]);
    atomicAdd(&orow[lane + 32], w * hr[lane + 32]);
}

__global__ void gat_elu(float* __restrict__ out, int total)
{
    const int i = blockIdx.x * blockDim.x + threadIdx.x;
    if (i < total) {
        const float v = out[i];
        out[i] = v > 0.f ? v : expm1f(v);
    }
}

// ---------------------------------------------------------------------------
extern "C" void kernel_launch(void* const* d_in, const int* in_sizes, int n_in,
                              void* d_out, int out_size, void* d_ws, size_t ws_size,
                              hipStream_t stream)
{
    const float* x    = (const float*)d_in[0];
    const int*   edge = (const int*)  d_in[1];
    const float* W    = (const float*)d_in[2];
    const float* attn = (const float*)d_in[3];

    const int N = in_sizes[0] / IN_F;
    const int E = in_sizes[1] / 2;
    const int* src = edge;          // edge[0,:]
    const int* dst = edge + E;      // edge[1,:]
    float* out = (float*)d_out;

    // Workspace layout: h | s_src | s_dst | m | denom  (N*68 floats = 27.2 MB)
    float* h     = (float*)d_ws;
    float* s_src = h + (size_t)N * OUT_F;
    float* s_dst = s_src + N;
    float* m     = s_dst + N;
    float* denom = m + N;

    const int totalNF = N * OUT_F;

    gat_init     <<<(totalNF + 255) / 256, 256, 0, stream>>>(out, m, denom, N);
    gat_gemm_wmma<<<(N + 127) / 128,       256, 0, stream>>>(x, W, h, N);
    gat_scores   <<<(N * 32 + 255) / 256,  256, 0, stream>>>(h, attn, s_src, s_dst, N);
    gat_edge_max <<<(E + 255) / 256,       256, 0, stream>>>(src, dst, s_src, s_dst, m, E);
    gat_edge_sum <<<(E + 255) / 256,       256, 0, stream>>>(src, dst, s_src, s_dst, m, denom, E);
    {
        const long long threads = (long long)E * 32;
        gat_edge_aggr<<<(int)((threads + 255) / 256), 256, 0, stream>>>(
            src, dst, s_src, s_dst, m, denom, h, out, E);
    }
    gat_elu      <<<(totalNF + 255) / 256, 256, 0, stream>>>(out, totalNF);
}